// Demolition_Conv2d_43439299231859
// MI455X (gfx1250) — compile-verified
//
#include <hip/hip_runtime.h>

typedef __attribute__((ext_vector_type(2))) float v2f;
typedef __attribute__((ext_vector_type(8))) float v8f;

#define Q_SCALE 1.6666666666666667f   /* 15/9 */
#define Q_INV   0.6f                  /* 9/15 */

// D(16 couts x 16 pixels) = A(16 couts x K) * B(K x 16 pixels), K = 9 taps + bias col, padded to 12.
// 3 chained V_WMMA_F32_16X16X4_F32 per input channel; quantize-dequant between channel GEMMs.
__global__ __launch_bounds__(256) void demolition_conv_wmma(
    const float* __restrict__ x,     // [8,16,112,112]
    const float* __restrict__ wgt,   // [16,32,1,3,3]
    const float* __restrict__ bias,  // [16,32]
    float* __restrict__ out)         // [8,32,112,112]
{
    constexpr int H = 112, Wd = 112, Cin = 16, Cout = 32, WT = 7; // 7 tiles of 16 along W

    const int lane = threadIdx.x & 31;
    const int wave = blockIdx.x * (blockDim.x >> 5) + (threadIdx.x >> 5);

    const bool lo = lane < 16;      // low half-wave: K = 4s,4s+1 ; high: K = 4s+2,4s+3
    const int  m  = lane & 15;      // A: cout row index | B: pixel column index

    // wave -> (cout half, b, h, w-tile)
    const int couthalf = wave & 1;
    int tile = wave >> 1;
    const int wtile = tile % WT;  tile /= WT;
    const int h = tile % H;
    const int b = tile / H;
    const int w0 = wtile * 16;
    const int coutbase = couthalf * 16;

    const int pw = w0 + m;                                   // this lane's pixel column (B matrix N)
    const float* wrow = wgt + (size_t)(coutbase + m) * 9;    // this lane's cout row (A matrix M)

    const int ta0 = lo ? 0 : 2;   // taps covered by this lane in WMMA step 0
    const int ta1 = lo ? 4 : 6;   // taps covered in step 1

    v8f acc = {};  // sum over Cin of quantize-dequant(per-channel conv)

    for (int cin = 0; cin < Cin; ++cin) {
        const float* xc = x + ((size_t)b * Cin + cin) * (size_t)(H * Wd);
        const float* wr = wrow + (size_t)cin * (Cout * 9);

        // ---- A matrix (weights), per-lane 2 values per K-step -------------
        v2f a0, a1, a2;
        a0[0] = wr[ta0];      a0[1] = wr[ta0 + 1];
        a1[0] = wr[ta1];      a1[1] = wr[ta1 + 1];
        a2[0] = lo ? wr[8] : 0.0f;                                   // K=8: tap 8   (hi: K=10 -> 0)
        a2[1] = lo ? bias[cin * Cout + coutbase + m] : 0.0f;         // K=9: bias    (hi: K=11 -> 0)

        // ---- B matrix (x 3x3 patch taps for this lane's pixel) ------------
        auto ldx = [&](int t) -> float {
            int dh = t / 3;
            int dw = t - dh * 3;
            int hh = h + dh - 1;
            int ww = pw + dw - 1;
            return (hh >= 0 && hh < H && ww >= 0 && ww < Wd) ? xc[hh * Wd + ww] : 0.0f;
        };
        v2f b0, b1, b2;
        b0[0] = ldx(ta0);     b0[1] = ldx(ta0 + 1);
        b1[0] = ldx(ta1);     b1[1] = ldx(ta1 + 1);
        b2[0] = lo ? ldx(8) : 0.0f;          // K=8: tap 8
        b2[1] = lo ? 1.0f   : 0.0f;          // K=9: bias column of ones

        // ---- per-channel GEMM tile: y = A*B (+bias via K=9) ---------------
        v8f c = {};
        c = __builtin_amdgcn_wmma_f32_16x16x4_f32(false, a0, false, b0, (short)0, c, false, false);
        c = __builtin_amdgcn_wmma_f32_16x16x4_f32(false, a1, false, b1, (short)0, c, false, false);
        c = __builtin_amdgcn_wmma_f32_16x16x4_f32(false, a2, false, b2, (short)0, c, false, false);

        // ---- STE quantize-dequant (RNE, matches jnp.round), then sum ------
#pragma unroll
        for (int j = 0; j < 8; ++j)
            acc[j] += rintf(c[j] * Q_SCALE) * Q_INV;
    }

    // D layout: VGPR j, lanes 0-15 -> (cout = coutbase+j,   pixel = w0+m)
    //           VGPR j, lanes16-31 -> (cout = coutbase+j+8, pixel = w0+m)
    const size_t obase =
        (((size_t)b * Cout + coutbase + (lo ? 0 : 8)) * H + h) * (size_t)Wd + w0 + m;
#pragma unroll
    for (int j = 0; j < 8; ++j)
        out[obase + (size_t)j * (H * Wd)] = acc[j];
}

extern "C" void kernel_launch(void* const* d_in, const int* in_sizes, int n_in,
                              void* d_out, int out_size, void* d_ws, size_t ws_size,
                              hipStream_t stream) {
    (void)in_sizes; (void)n_in; (void)out_size; (void)d_ws; (void)ws_size;
    const float* x    = (const float*)d_in[0];   // [8,16,112,112]
    const float* wgt  = (const float*)d_in[1];   // [16,32,1,3,3]
    const float* bias = (const float*)d_in[2];   // [16,32]
    float* out = (float*)d_out;                  // [8,32,112,112]

    // waves = 8 * 112 * 7 pixel-tiles * 2 cout-halves = 12544 ; 8 waves / 256-thread block
    const int waves  = 8 * 112 * 7 * 2;
    const int blocks = waves / 8;                // 1568
    demolition_conv_wmma<<<blocks, 256, 0, stream>>>(x, wgt, bias, out);
}